// CopyRNN_53979148976578
// MI455X (gfx1250) — compile-verified
//
#include <hip/hip_runtime.h>
#include <cstdint>
#include <cstddef>

// Problem constants (from the reference)
#define B_    256
#define S_    512
#define SRC_  512
#define TH_   512
#define E_    128
#define V_    50000
#define OOV_  100
#define VO_   50100   // V + OOV

typedef float v2f __attribute__((ext_vector_type(2)));
typedef float v8f __attribute__((ext_vector_type(8)));
typedef unsigned int v4u __attribute__((ext_vector_type(4)));
typedef int v4i __attribute__((ext_vector_type(4)));
typedef int v8i __attribute__((ext_vector_type(8)));

#if __has_builtin(__builtin_amdgcn_tensor_load_to_lds) && __has_builtin(__builtin_amdgcn_s_wait_tensorcnt)
#define USE_TDM 1
#else
#define USE_TDM 0
#endif

__device__ __forceinline__ float sigf(float x) { return 1.0f / (1.0f + expf(-x)); }

#if USE_TDM
// ---------------------------------------------------------------------------
// Tensor Data Mover: 2D tile load global->LDS, D# built per CDNA5 ISA ch.8.
//   data_size=4B; optional LDS padding (pad_amount_enc<0 disables).
//   Rows >= tensor_d1 are zero-filled by TDM OOB semantics.
// Issue from ONE wave only; retire via s_wait_tensorcnt + workgroup barrier.
// ---------------------------------------------------------------------------
__device__ __forceinline__ void tdm_load_2d(
    unsigned int lds_off, const float* gptr,
    int tensor_d0, int tensor_d1, int tile_d0, int tile_d1,
    int row_stride_elems, int pad_interval_enc, int pad_amount_enc)
{
  const unsigned long long ga = (unsigned long long)(uintptr_t)gptr;
  v4u g0;
  g0[0] = 1u;                                   // count=1, user descriptor
  g0[1] = lds_off;                              // lds_addr (bytes)
  g0[2] = (unsigned int)ga;                     // global_addr[31:0]
  g0[3] = (unsigned int)((ga >> 32) & 0x01FFFFFFu) | (2u << 30);  // addr[56:32] | type=2
  unsigned int w0 = (2u << 16);                 // data_size=4B
  if (pad_amount_enc >= 0)
    w0 |= (1u << 20) | ((unsigned int)pad_interval_enc << 22) | ((unsigned int)pad_amount_enc << 25);
  v8i g1;
  g1[0] = (int)w0;
  g1[1] = (int)((unsigned int)(tensor_d0 & 0xFFFF) << 16);                       // tensor_dim0 lo16
  g1[2] = (int)(((unsigned int)tensor_d0 >> 16) |
                ((unsigned int)(tensor_d1 & 0xFFFF) << 16));                     // d0 hi | d1 lo
  g1[3] = (int)(((unsigned int)tensor_d1 >> 16) | ((unsigned int)tile_d0 << 16));// d1 hi | tile_dim0
  g1[4] = (int)(unsigned int)tile_d1;                                            // tile_dim1 (dim2=0)
  g1[5] = (int)(unsigned int)row_stride_elems;                                   // tensor_dim0_stride lo32
  g1[6] = 0;
  g1[7] = 0;
  v4i z4 = {0, 0, 0, 0};
#if defined(__clang_major__) && __clang_major__ >= 23
  v8i z8 = {0, 0, 0, 0, 0, 0, 0, 0};
  __builtin_amdgcn_tensor_load_to_lds(g0, g1, z4, z4, z8, 0);
#else
  __builtin_amdgcn_tensor_load_to_lds(g0, g1, z4, z4, 0);
#endif
}
#endif

// ---------------------------------------------------------------------------
// Generic f32 WMMA GEMM:  C[M][N] = A[M][K] * W[N][K]^T  (+bias / act per mode)
// Block tile 32(M) x 64(N), K-chunk 32, 8 waves (2x4 grid of 16x16 WMMA tiles).
// modes: 0=store  1=+bias1(+bias2)  2=accumulate into C  3=tanh(+bias1)  4=exp
// Rows of W with index >= nwValid are treated as zero (gen-GEMM OOV padding).
// TDM path: double-buffered tiles; DMA of chunk i+1 overlaps WMMA of chunk i
// (s_wait_tensorcnt retires exactly the previous chunk, TDM is in-order).
// Pad 4 DWORDs every 32 -> LDS row stride 36, conflict-free f32x2 lane reads.
// ---------------------------------------------------------------------------
__global__ __launch_bounds__(256) void k_gemm(
    const float* __restrict__ A, int lda,
    const float* __restrict__ W, int ldw, int nwValid,
    float* __restrict__ C, int ldc,
    const float* __restrict__ bias1, const float* __restrict__ bias2,
    int M, int N, int K, int mode)
{
  __shared__ float As[2][32][36];
  __shared__ float Ws[2][64][36];

  const int tid   = threadIdx.x;
  const int lane  = tid & 31;
  const int wid   = tid >> 5;
  const int waveM = wid >> 2;    // 0..1
  const int waveN = wid & 3;     // 0..3
  const int m0    = blockIdx.y * 32;
  const int n0    = blockIdx.x * 64;

  v8f acc = {0.f,0.f,0.f,0.f,0.f,0.f,0.f,0.f};

  // ISA layouts, V_WMMA_F32_16X16X4_F32:
  //  A 16x4 : vgpr j, lane L -> A[L%16][j + (L/16)*2]
  //  B 4x16 : vgpr j, lane L -> B[j + (L/16)*2][L%16]   (B[k][n] = W[n][k])
  const int rA = waveM * 16 + (lane & 15);
  const int rB = waveN * 16 + (lane & 15);
  const int ko = (lane >> 4) << 1;

  int wrows = nwValid - n0;              // valid W rows in this tile
  if (wrows > 64) wrows = 64;

#if USE_TDM
  const int nchunks = K >> 5;
  const unsigned int offA0 = (unsigned int)(uintptr_t)&As[0][0][0];
  const unsigned int offA1 = (unsigned int)(uintptr_t)&As[1][0][0];
  const unsigned int offW0 = (unsigned int)(uintptr_t)&Ws[0][0][0];
  const unsigned int offW1 = (unsigned int)(uintptr_t)&Ws[1][0][0];

  if (wrows <= 0) {                      // fully OOV tail tile: zero W once
    for (int i = tid; i < 2 * 64 * 36; i += 256) (&Ws[0][0][0])[i] = 0.f;
  }
  if (tid < 32) {                        // wave 0 drives the Tensor Data Mover
    tdm_load_2d(offA0, A + (size_t)m0 * lda, 32, 32, 32, 32, lda, 4, 3);
    if (wrows > 0)
      tdm_load_2d(offW0, W + (size_t)n0 * ldw, 32, wrows, 32, 64, ldw, 4, 3);
  }
  for (int ic = 0; ic < nchunks; ++ic) {
    const int buf = ic & 1;
    if (tid < 32) {
      if (ic + 1 < nchunks) {            // prefetch next chunk into other buffer
        const int kc = (ic + 1) << 5;
        tdm_load_2d(buf ? offA0 : offA1, A + (size_t)m0 * lda + kc, 32, 32, 32, 32, lda, 4, 3);
        if (wrows > 0) {
          tdm_load_2d(buf ? offW0 : offW1, W + (size_t)n0 * ldw + kc, 32, wrows, 32, 64, ldw, 4, 3);
          __builtin_amdgcn_s_wait_tensorcnt(2);   // chunk ic retired, ic+1 in flight
        } else {
          __builtin_amdgcn_s_wait_tensorcnt(1);
        }
      } else {
        __builtin_amdgcn_s_wait_tensorcnt(0);
      }
    }
    __syncthreads();
#pragma unroll
    for (int kk = 0; kk < 32; kk += 4) {
      v2f a, b;
      a[0] = As[buf][rA][kk + ko]; a[1] = As[buf][rA][kk + ko + 1];
      b[0] = Ws[buf][rB][kk + ko]; b[1] = Ws[buf][rB][kk + ko + 1];
      acc = __builtin_amdgcn_wmma_f32_16x16x4_f32(false, a, false, b, (short)0, acc, false, false);
    }
    __syncthreads();                     // before TDM may overwrite this buffer
  }
#else
  const int lr = tid >> 3;               // 0..31
  const int lc = (tid & 7) << 2;         // 0..28
  for (int kc = 0; kc < K; kc += 32) {
    {
      const float4 v = *reinterpret_cast<const float4*>(A + (size_t)(m0 + lr) * lda + kc + lc);
      As[0][lr][lc] = v.x; As[0][lr][lc+1] = v.y; As[0][lr][lc+2] = v.z; As[0][lr][lc+3] = v.w;
    }
#pragma unroll
    for (int h = 0; h < 2; ++h) {
      const int r  = lr + h * 32;
      const int gr = n0 + r;
      float4 v = make_float4(0.f, 0.f, 0.f, 0.f);
      if (gr < nwValid) v = *reinterpret_cast<const float4*>(W + (size_t)gr * ldw + kc + lc);
      Ws[0][r][lc] = v.x; Ws[0][r][lc+1] = v.y; Ws[0][r][lc+2] = v.z; Ws[0][r][lc+3] = v.w;
    }
    __syncthreads();
#pragma unroll
    for (int kk = 0; kk < 32; kk += 4) {
      v2f a, b;
      a[0] = As[0][rA][kk + ko]; a[1] = As[0][rA][kk + ko + 1];
      b[0] = Ws[0][rB][kk + ko]; b[1] = Ws[0][rB][kk + ko + 1];
      acc = __builtin_amdgcn_wmma_f32_16x16x4_f32(false, a, false, b, (short)0, acc, false, false);
    }
    __syncthreads();
  }
#endif

  // Epilogue. C/D layout: vgpr r, lane L -> C[r + (L/16)*8][L%16]
  const int nn = n0 + waveN * 16 + (lane & 15);
  if (nn >= N) return;
  const int mb = m0 + waveM * 16 + ((lane >> 4) << 3);
  float badd = 0.f;
  if (mode == 1 || mode == 3) {
    if (bias1) badd = bias1[nn];
    if (bias2) badd += bias2[nn];
  }
#pragma unroll
  for (int r = 0; r < 8; ++r) {
    const float v = acc[r];
    float* p = C + (size_t)(mb + r) * ldc + nn;
    if      (mode == 0) *p = v;
    else if (mode == 1) *p = v + badd;
    else if (mode == 2) *p = *p + v;
    else if (mode == 3) *p = tanhf(v + badd);
    else                *p = expf(v);
  }
}

// ---------------------------------------------------------------------------
// copy_state[b,:] = mean over non-matching s of enc[b,s,:]
// (in_copy softmax collapses to uniform over non-match positions; W_incopy dead)
// ---------------------------------------------------------------------------
__global__ __launch_bounds__(256) void k_copy_state(
    const float* __restrict__ enc, const int* __restrict__ ptok,
    const int* __restrict__ stok, float* __restrict__ cs)
{
  __shared__ unsigned char msh[S_];
  __shared__ int cnt;
  const int b = blockIdx.x, tid = threadIdx.x;
  const int pt = ptok[b];
  if (tid == 0) cnt = 0;
  __syncthreads();
  int loc = 0;
  for (int i = tid; i < S_; i += 256) {
    const int mm = (stok[(size_t)b * S_ + i] == pt) ? 1 : 0;
    msh[i] = (unsigned char)mm; loc += mm;
  }
  atomicAdd(&cnt, loc);
  __syncthreads();
  const float inv = 1.0f / (float)(S_ - cnt);
  const float* base = enc + (size_t)b * S_ * SRC_;
  float a0 = 0.f, a1 = 0.f;
  for (int s = 0; s < S_; ++s) {
    if (!msh[s]) { a0 += base[(size_t)s * SRC_ + tid]; a1 += base[(size_t)s * SRC_ + tid + 256]; }
  }
  cs[(size_t)b * SRC_ + tid] = a0 * inv;
  cs[(size_t)b * SRC_ + tid + 256] = a1 * inv;
}

// x = [embedding[tok], prev_context_state, copy_state]  (B x 1152)
__global__ __launch_bounds__(256) void k_build_x(
    const float* __restrict__ emb, const int* __restrict__ ptok,
    const float* __restrict__ pcs, const float* __restrict__ cs, float* __restrict__ x)
{
  const int b = blockIdx.x, tid = threadIdx.x;
  int pt = ptok[b];
  if (pt >= V_) pt = 1;  // OOV -> UNK
  float* row = x + (size_t)b * 1152;
  if (tid < E_) row[tid] = emb[(size_t)pt * E_ + tid];
  for (int i = tid; i < 512; i += 256) {
    row[128 + i] = pcs[(size_t)b * TH_ + i];
    row[640 + i] = cs[(size_t)b * SRC_ + i];
  }
}

// LSTM gate activations -> h1, c1 (written straight into d_out)
__global__ __launch_bounds__(256) void k_lstm_act(
    const float* __restrict__ gates, const float* __restrict__ c0,
    float* __restrict__ h1, float* __restrict__ c1)
{
  const int b = blockIdx.x, tid = threadIdx.x;
  const float* g = gates + (size_t)b * 2048;
  for (int j = tid; j < TH_; j += 256) {
    const float ig = g[j], fg = g[512 + j], gg = g[1024 + j], og = g[1536 + j];
    const float c = sigf(fg) * c0[(size_t)b * TH_ + j] + sigf(ig) * tanhf(gg);
    const float h = sigf(og) * tanhf(c);
    c1[(size_t)b * TH_ + j] = c;
    h1[(size_t)b * TH_ + j] = h;
  }
}

// attn_logits[b,s] = q[b,:] . enc[b,s,:]   (masked -> -inf); wave-per-row
__global__ __launch_bounds__(256) void k_attn_logits(
    const float* __restrict__ q, const float* __restrict__ enc,
    const unsigned char* __restrict__ pmask, float* __restrict__ lg)
{
  __shared__ float qs[SRC_];
  const int b = blockIdx.y, tid = threadIdx.x, lane = tid & 31, wid = tid >> 5;
  for (int i = tid; i < SRC_; i += 256) qs[i] = q[(size_t)b * SRC_ + i];
  __syncthreads();
  const int sbase = blockIdx.x * 64 + wid * 8;
  for (int rr = 0; rr < 8; ++rr) {
    const int s = sbase + rr;
    const float* row = enc + ((size_t)b * S_ + s) * SRC_;
    float acc = 0.f;
    for (int it = 0; it < 16; ++it) { const int d = it * 32 + lane; acc += qs[d] * row[d]; }
#pragma unroll
    for (int off = 16; off > 0; off >>= 1) acc += __shfl_xor(acc, off, 32);
    if (lane == 0)
      lg[(size_t)b * S_ + s] = pmask[(size_t)b * S_ + s] ? -__builtin_huge_valf() : acc;
  }
}

__global__ __launch_bounds__(256) void k_softmax(float* __restrict__ lg)
{
  __shared__ float red[256];
  const int b = blockIdx.x, tid = threadIdx.x;
  const float v0 = lg[(size_t)b * S_ + tid], v1 = lg[(size_t)b * S_ + tid + 256];
  red[tid] = fmaxf(v0, v1); __syncthreads();
  for (int o = 128; o > 0; o >>= 1) { if (tid < o) red[tid] = fmaxf(red[tid], red[tid + o]); __syncthreads(); }
  const float m = red[0]; __syncthreads();
  const float e0 = expf(v0 - m), e1 = expf(v1 - m);
  red[tid] = e0 + e1; __syncthreads();
  for (int o = 128; o > 0; o >>= 1) { if (tid < o) red[tid] += red[tid + o]; __syncthreads(); }
  const float inv = 1.0f / red[0];
  lg[(size_t)b * S_ + tid] = e0 * inv;
  lg[(size_t)b * S_ + tid + 256] = e1 * inv;
}

// context[b,:] = sum_s attn_w[b,s] * enc[b,s,:]  (coalesced streaming pass)
__global__ __launch_bounds__(256) void k_context(
    const float* __restrict__ w, const float* __restrict__ enc, float* __restrict__ ctx)
{
  __shared__ float wsh[S_];
  const int b = blockIdx.x, tid = threadIdx.x;
  for (int i = tid; i < S_; i += 256) wsh[i] = w[(size_t)b * S_ + i];
  __syncthreads();
  const float* base = enc + (size_t)b * S_ * SRC_;
  float a0 = 0.f, a1 = 0.f;
  for (int s = 0; s < S_; ++s) {
    const float wv = wsh[s];
    a0 += wv * base[(size_t)s * SRC_ + tid];
    a1 += wv * base[(size_t)s * SRC_ + tid + 256];
  }
  ctx[(size_t)b * SRC_ + tid] = a0;
  ctx[(size_t)b * SRC_ + tid + 256] = a1;
}

// xo = [context, h1]  (B x 1024)
__global__ __launch_bounds__(256) void k_build_xo(
    const float* __restrict__ ctx, const float* __restrict__ h1, float* __restrict__ xo)
{
  const int b = blockIdx.x, tid = threadIdx.x;
  float* row = xo + (size_t)b * 1024;
  for (int i = tid; i < 512; i += 256) {
    row[i] = ctx[(size_t)b * SRC_ + i];
    row[512 + i] = h1[(size_t)b * TH_ + i];
  }
}

// ---------------------------------------------------------------------------
// Copy branch: 32 enc rows (full K=512) resident in dynamic LDS via one TDM
// load (pad 2 DWORDs every 256 -> row stride 516, conflict-free). The 128
// W_copy K-chunks (8 h-tiles x 16 chunks) stream through a 2-deep TDM
// pipeline; each 16x16 result gets tanh, dot with attn_out[b], then exp +
// atomic scatter into the total distribution.
// Dynamic LDS: Ae 32*516 + Bs 2*64*36 floats = 84480 bytes.
// ---------------------------------------------------------------------------
__device__ __forceinline__ int aeidx(int r, int c) {
  return r * 516 + c + ((c >> 8) << 1);   // +2 pad after each 256 DWORDs
}

__global__ __launch_bounds__(256) void k_copy(
    const float* __restrict__ enc, const float* __restrict__ Wc,
    const float* __restrict__ attn, const int* __restrict__ stok,
    const unsigned char* __restrict__ pmask, float* __restrict__ total)
{
  extern __shared__ float smem[];
  float* Ae  = smem;                 // 32 rows * 516
  float* Bs0 = smem + 32 * 516;      // 64 rows * 36
  float* Bs1 = Bs0 + 64 * 36;        // second buffer (TDM pipeline)
  __shared__ float avS[TH_];
  __shared__ float psS[32];

  const int b  = blockIdx.y;
  const int s0 = blockIdx.x * 32;
  const int tid = threadIdx.x, lane = tid & 31, wid = tid >> 5;
  const int waveM = wid >> 2;       // 0..1 (s sub-tile)
  const int waveN = wid & 3;        // 0..3 (h sub-tile)

  for (int i = tid; i < TH_; i += 256) avS[i] = attn[(size_t)b * TH_ + i];
  if (tid < 32) psS[tid] = 0.f;

  const int rA = waveM * 16 + (lane & 15);
  const int rB = waveN * 16 + (lane & 15);
  const int ko = (lane >> 4) << 1;

  float part[8];
#pragma unroll
  for (int r = 0; r < 8; ++r) part[r] = 0.f;

  v8f acc = {0.f,0.f,0.f,0.f,0.f,0.f,0.f,0.f};

#if USE_TDM
  const unsigned int offB0 = (unsigned int)(uintptr_t)Bs0;
  const unsigned int offB1 = (unsigned int)(uintptr_t)Bs1;
  if (tid < 32) {
    tdm_load_2d((unsigned int)(uintptr_t)Ae, enc + ((size_t)b * S_ + s0) * SRC_,
                512, 32, 512, 32, SRC_, 7, 1);                 // enc tile, padded rows
    tdm_load_2d(offB0, Wc, 32, 64, 32, 64, SRC_, 4, 3);        // chunk 0 (nt=0,kc=0)
  }
  for (int c = 0; c < 128; ++c) {       // chunk c: nt = c>>4 (h-tile), kc = (c&15)*32
    const int buf = c & 1;
    if (tid < 32) {
      if (c + 1 < 128) {
        const int nt1 = (c + 1) >> 4, kc1 = ((c + 1) & 15) << 5;
        tdm_load_2d(buf ? offB0 : offB1, Wc + (size_t)(nt1 * 64) * SRC_ + kc1,
                    32, 64, 32, 64, SRC_, 4, 3);
        __builtin_amdgcn_s_wait_tensorcnt(1);   // chunk c (and enc tile) retired
      } else {
        __builtin_amdgcn_s_wait_tensorcnt(0);
      }
    }
    __syncthreads();
    const int kc = (c & 15) << 5;
    if ((c & 15) == 0) {
      v8f z = {0.f,0.f,0.f,0.f,0.f,0.f,0.f,0.f};
      acc = z;
    }
    const float* Bsb = buf ? Bs1 : Bs0;
#pragma unroll
    for (int kk = 0; kk < 32; kk += 4) {
      v2f a, bb;
      const int ai = aeidx(rA, kc + kk + ko);
      a[0]  = Ae[ai];                 a[1]  = Ae[ai + 1];
      bb[0] = Bsb[rB * 36 + kk + ko]; bb[1] = Bsb[rB * 36 + kk + ko + 1];
      acc = __builtin_amdgcn_wmma_f32_16x16x4_f32(false, a, false, bb, (short)0, acc, false, false);
    }
    if ((c & 15) == 15) {
      const int n0 = (c >> 4) * 64;
      const float aw = avS[n0 + waveN * 16 + (lane & 15)];
#pragma unroll
      for (int r = 0; r < 8; ++r) part[r] += tanhf(acc[r]) * aw;
    }
    __syncthreads();                   // before TDM may overwrite this buffer
  }
#else
  const int lr = tid >> 3, lc = (tid & 7) << 2;
  for (int i = tid; i < 32 * 128; i += 256) {
    const int r = i >> 7, c = (i & 127) << 2;
    const float4 v = *reinterpret_cast<const float4*>(enc + ((size_t)b * S_ + (s0 + r)) * SRC_ + c);
    const int o = aeidx(r, c);
    Ae[o] = v.x; Ae[o+1] = v.y; Ae[o+2] = v.z; Ae[o+3] = v.w;
  }
  __syncthreads();
  for (int nt = 0; nt < 8; ++nt) {             // h-tiles of 64
    const int n0 = nt * 64;
    v8f z = {0.f,0.f,0.f,0.f,0.f,0.f,0.f,0.f};
    acc = z;
    for (int kc = 0; kc < SRC_; kc += 32) {
      __syncthreads();
#pragma unroll
      for (int h = 0; h < 2; ++h) {
        const int r = lr + h * 32;
        const float4 v = *reinterpret_cast<const float4*>(Wc + (size_t)(n0 + r) * SRC_ + kc + lc);
        float* p = Bs0 + r * 36 + lc;
        p[0] = v.x; p[1] = v.y; p[2] = v.z; p[3] = v.w;
      }
      __syncthreads();
#pragma unroll
      for (int kk = 0; kk < 32; kk += 4) {
        v2f a, bb;
        const int ai = aeidx(rA, kc + kk + ko);
        a[0]  = Ae[ai];                 a[1]  = Ae[ai + 1];
        bb[0] = Bs0[rB * 36 + kk + ko]; bb[1] = Bs0[rB * 36 + kk + ko + 1];
        acc = __builtin_amdgcn_wmma_f32_16x16x4_f32(false, a, false, bb, (short)0, acc, false, false);
      }
    }
    const float aw = avS[n0 + waveN * 16 + (lane & 15)];
#pragma unroll
    for (int r = 0; r < 8; ++r) part[r] += tanhf(acc[r]) * aw;
  }
#endif

  const int sl = waveM * 16 + ((lane >> 4) << 3);   // row base within tile
#pragma unroll
  for (int r = 0; r < 8; ++r) atomicAdd(&psS[sl + r], part[r]);
  __syncthreads();

  if (tid < 32) {
    const int s = s0 + tid;
    if (!pmask[(size_t)b * S_ + s]) {            // masked -> exp(-inf)=0: skip
      const float v = expf(psS[tid]);
      const int t = stok[(size_t)b * S_ + s];
      unsafeAtomicAdd(total + (size_t)b * VO_ + t, v);
    }
  }
}

// total_prob = log(total / rowsum)
__global__ __launch_bounds__(256) void k_norm(float* __restrict__ out)
{
  __shared__ float red[256];
  const int b = blockIdx.x, tid = threadIdx.x;
  float* row = out + (size_t)b * VO_;
  float s = 0.f;
  for (int i = tid; i < VO_; i += 256) s += row[i];
  red[tid] = s; __syncthreads();
  for (int o = 128; o > 0; o >>= 1) { if (tid < o) red[tid] += red[tid + o]; __syncthreads(); }
  const float ls = logf(red[0]);
  for (int i = tid; i < VO_; i += 256) row[i] = logf(row[i]) - ls;
}

// ---------------------------------------------------------------------------
extern "C" void kernel_launch(void* const* d_in, const int* in_sizes, int n_in,
                              void* d_out, int out_size, void* d_ws, size_t ws_size,
                              hipStream_t stream)
{
  (void)in_sizes; (void)n_in; (void)out_size; (void)ws_size;
  const float* emb    = (const float*)d_in[0];
  const float* W_ih   = (const float*)d_in[1];
  const float* W_hh   = (const float*)d_in[2];
  const float* b_ih   = (const float*)d_in[3];
  const float* b_hh   = (const float*)d_in[4];
  const float* W_attn = (const float*)d_in[5];
  const float* W_out  = (const float*)d_in[6];
  const float* b_out  = (const float*)d_in[7];
  const float* W_copy = (const float*)d_in[8];
  // d_in[9] = W_incopy: provably unused (masked to -inf / exact zeros)
  const float* W_gen  = (const float*)d_in[10];
  const float* enc    = (const float*)d_in[11];
  const float* pcs    = (const float*)d_in[12];
  const float* h0     = (const float*)d_in[13];
  const float* c0     = (const float*)d_in[14];
  const int*   ptok   = (const int*)d_in[15];
  const int*   stok   = (const int*)d_in[16];
  const unsigned char* pmask = (const unsigned char*)d_in[17];

  float* out   = (float*)d_out;
  float* total = out;                                  // B x VO_
  float* attn  = out + (size_t)B_ * VO_;               // B x TH_
  float* h1    = attn + (size_t)B_ * TH_;              // B x TH_
  float* c1    = h1 + (size_t)B_ * TH_;                // B x TH_

  float* ws    = (float*)d_ws;
  float* cs    = ws;               // B x 512
  float* x     = cs + 131072;      // B x 1152
  float* gates = x + 294912;       // B x 2048
  float* q     = gates + 524288;   // B x 512
  float* lg    = q + 131072;       // B x 512 (logits -> weights in place)
  float* ctx   = lg + 131072;      // B x 512
  float* xo    = ctx + 131072;     // B x 1024

  // Dynamic LDS for k_copy: sized for the TDM double-buffer path regardless of
  // host-side preprocessor state (host pass cannot see device __has_builtin).
  const size_t copy_lds = (size_t)(32 * 516 + 2 * 64 * 36) * sizeof(float);

  // 1) copy_state (uniform mean over non-match positions)
  k_copy_state<<<B_, 256, 0, stream>>>(enc, ptok, stok, cs);
  // 2) LSTM input concat
  k_build_x<<<B_, 256, 0, stream>>>(emb, ptok, pcs, cs, x);
  // 3) gates = x @ W_ih^T + (b_ih + b_hh);  4) gates += h0 @ W_hh^T
  k_gemm<<<dim3(32, 8), 256, 0, stream>>>(x, 1152, W_ih, 1152, 2048, gates, 2048, b_ih, b_hh, 256, 2048, 1152, 1);
  k_gemm<<<dim3(32, 8), 256, 0, stream>>>(h0, 512, W_hh, 512, 2048, gates, 2048, nullptr, nullptr, 256, 2048, 512, 2);
  // 5) LSTM activations -> h1, c1 (into d_out)
  k_lstm_act<<<B_, 256, 0, stream>>>(gates, c0, h1, c1);
  // 6) q = h1 @ W_attn^T
  k_gemm<<<dim3(8, 8), 256, 0, stream>>>(h1, 512, W_attn, 512, 512, q, 512, nullptr, nullptr, 256, 512, 512, 0);
  // 7-9) attention: logits -> softmax -> context
  k_attn_logits<<<dim3(8, B_), 256, 0, stream>>>(q, enc, pmask, lg);
  k_softmax<<<B_, 256, 0, stream>>>(lg);
  k_context<<<B_, 256, 0, stream>>>(lg, enc, ctx);
  // 10-11) attn_out = tanh([context,h1] @ W_out^T + b_out) (into d_out)
  k_build_xo<<<B_, 256, 0, stream>>>(ctx, h1, xo);
  k_gemm<<<dim3(8, 8), 256, 0, stream>>>(xo, 1024, W_out, 1024, 512, attn, 512, b_out, nullptr, 256, 512, 1024, 3);
  // 12) total = exp(attn_out @ W_gen^T), cols >= V get exp(0)=1 via zero-padded W rows
  k_gemm<<<dim3((VO_ + 63) / 64, 8), 256, 0, stream>>>(attn, 512, W_gen, 512, V_, total, VO_, nullptr, nullptr, 256, VO_, 512, 4);
  // 13) copy branch: total[b, stok] += exp(copy_seq) (pipelined TDM + WMMA + scatter)
  k_copy<<<dim3(S_ / 32, B_), 256, copy_lds, stream>>>(enc, W_copy, attn, stok, pmask, total);
  // 14) log-normalize rows
  k_norm<<<B_, 256, 0, stream>>>(total);
}